// MyModel_20555713479331
// MI455X (gfx1250) — compile-verified
//
#include <hip/hip_runtime.h>

// ---------------- problem constants (from reference) ----------------
#define NNODES  2500
#define KB      4
#define NEDGES  40000
#define NLAYERS 10
#define BATCH   64
#define VSZ     (NNODES * KB)          // 10000
#define INV_STD 0.9995003746879195f    // 1/sqrt(1 + 1e-3)

typedef __attribute__((ext_vector_type(2))) float v2f;
typedef __attribute__((ext_vector_type(8))) float v8f;

// Build CSR row pointers from the sorted COO row array (lower_bound per node).
__global__ void rowptr_kernel(const int* __restrict__ rows, int* __restrict__ row_ptr) {
    int i = blockIdx.x * blockDim.x + threadIdx.x;
    if (i > NNODES) return;
    int lo = 0, hi = NEDGES;
    while (lo < hi) {
        int mid = (lo + hi) >> 1;
        if (rows[mid] < i) lo = mid + 1; else hi = mid;
    }
    row_ptr[i] = lo;
}

// h stored FEATURE-MAJOR: h[v * BATCH + b].
// h[v,b] = sum_k x[b,k] * W1[k,v] + b1[v]
__global__ void init_dense_kernel(const float* __restrict__ x, const float* __restrict__ W1,
                                  const float* __restrict__ b1, float* __restrict__ h) {
    int idx = blockIdx.x * blockDim.x + threadIdx.x;
    if (idx >= BATCH * VSZ) return;
    int b = idx & (BATCH - 1);
    int v = idx >> 6;
    float t = b1[v];
    t = fmaf(x[b * 4 + 0], W1[0 * VSZ + v], t);
    t = fmaf(x[b * 4 + 1], W1[1 * VSZ + v], t);
    t = fmaf(x[b * 4 + 2], W1[2 * VSZ + v], t);
    t = fmaf(x[b * 4 + 3], W1[3 * VSZ + v], t);
    h[idx] = t;
}

// PReLU -> BatchNorm(inference) -> optional low-rank residual (x @ W_rep)
// Both hin/hout feature-major [VSZ][BATCH]; fully coalesced, params uniform per 64 lanes.
__global__ void act_kernel(const float* __restrict__ hin, float* __restrict__ hout,
                           const float* __restrict__ alpha, const float* __restrict__ gam,
                           const float* __restrict__ beta,
                           const float* __restrict__ x, const float* __restrict__ Wr,
                           int has_res) {
    int idx = blockIdx.x * blockDim.x + threadIdx.x;
    if (idx >= BATCH * VSZ) return;
    int b = idx & (BATCH - 1);
    int v = idx >> 6;
    float t = hin[idx];
    t = fmaxf(t, 0.0f) + alpha[v] * fminf(t, 0.0f);        // PReLU
    t = fmaf(gam[v], t * INV_STD, beta[v]);                // BN (mean=0, var=1)
    if (has_res) {
        t = fmaf(x[b * 4 + 0], Wr[0 * VSZ + v], t);
        t = fmaf(x[b * 4 + 1], Wr[1 * VSZ + v], t);
        t = fmaf(x[b * 4 + 2], Wr[2 * VSZ + v], t);
        t = fmaf(x[b * 4 + 3], Wr[3 * VSZ + v], t);
    }
    hout[idx] = t;
}

// Block SpMM via V_WMMA_F32_16X16X4_F32.
// Block = one destination node, 4 waves = 4 batch tiles of 16 rows.
// Input h is feature-major [VSZ][BATCH] -> each edge gather is a contiguous 1KB
// region per workgroup (4 features x 64 batches).
// Per edge: A(16x4) = gathered h block (batch x src-feature),
//           B(4x16) = V_e^T in columns 0..3, zero elsewhere (cndmask, no EXEC div),
//           D(16x16 f32) accumulates over all edges of the node.
// TRANS_OUT=true  -> write feature-major [VSZ][BATCH] (intermediate layers, b128 stores)
// TRANS_OUT=false -> write batch-major [BATCH][VSZ]  (final layer, matches reference)
template <bool TRANS_OUT>
__global__ __launch_bounds__(128)
void spmm_wmma_kernel(const float* __restrict__ vals, const float* __restrict__ sbias,
                      const float* __restrict__ h, const int* __restrict__ row_ptr,
                      const int* __restrict__ cols, float* __restrict__ out) {
    const int node = blockIdx.x;
    const int lane = threadIdx.x & 31;
    const int b0   = (threadIdx.x >> 5) * 16;   // batch tile base (wave id * 16)
    const int m    = lane & 15;                 // A: batch row; B/D: column N
    const int kb   = (lane >> 4) * 2;           // K base for this lane half (A and B)
    const bool bvalid = (m < KB);

    // loop-invariant lane bases
    const float* __restrict__ hbase = h + b0 + m + (size_t)kb * BATCH;
    const float* __restrict__ vbase = vals + (m & 3) * KB + kb;  // safe alias for lanes m>=4

    v8f c = {};                                 // f32 16x16 accumulator, 8 VGPRs

    const int e0 = row_ptr[node];
    const int e1 = row_ptr[node + 1];
    for (int e = e0; e < e1; ++e) {
        const int cn = cols[e];                 // wave-uniform -> scalar load

        // A[m, kb], A[m, kb+1]: coalesced 64B segments per lane half
        const float* hp = hbase + (size_t)cn * (KB * BATCH);
        v2f a;
        a.x = hp[0];
        a.y = hp[BATCH];

        // B[kb, n] = V_e[n, kb] for n < 4 (V_e^T); streamed once -> non-temporal.
        // Unconditional load from aliased address, then cndmask to zero: no EXEC churn.
        v2f bb = __builtin_nontemporal_load(
                     reinterpret_cast<const v2f*>(vbase + (size_t)e * (KB * KB)));
        bb.x = bvalid ? bb.x : 0.0f;
        bb.y = bvalid ? bb.y : 0.0f;

        // D = A x B + C   (v_wmma_f32_16x16x4_f32)
        c = __builtin_amdgcn_wmma_f32_16x16x4_f32(
                /*neg_a=*/false, a, /*neg_b=*/false, bb,
                /*c_mod=*/(short)0, c, /*reuse_a=*/false, /*reuse_b=*/false);
    }

    // Writeback: lanes with N < 4 hold the real output columns.
    // VGPR r of lane L holds D[M = r + 8*(L>=16), N = L&15]; M is the batch index.
    if (bvalid) {
        const float sb = sbias[node * KB + m];
        const int mbase = (lane >> 4) * 8;
        if (TRANS_OUT) {
            // feature-major: row (node*4+m), batches b0+mbase .. +7 are contiguous
            float* op = out + (size_t)(node * KB + m) * BATCH + b0 + mbase;
            float4 lo = make_float4(c[0] + sb, c[1] + sb, c[2] + sb, c[3] + sb);
            float4 hi = make_float4(c[4] + sb, c[5] + sb, c[6] + sb, c[7] + sb);
            *reinterpret_cast<float4*>(op)     = lo;   // 32B-aligned
            *reinterpret_cast<float4*>(op + 4) = hi;
        } else {
            // batch-major reference layout [BATCH][VSZ]
            float* op = out + (size_t)node * KB + m;
#pragma unroll
            for (int r = 0; r < 8; ++r) {
                op[(size_t)(b0 + mbase + r) * VSZ] = c[r] + sb;
            }
        }
    }
}

extern "C" void kernel_launch(void* const* d_in, const int* in_sizes, int n_in,
                              void* d_out, int out_size, void* d_ws, size_t ws_size,
                              hipStream_t stream) {
    const float* x     = (const float*)d_in[0];
    const float* W1    = (const float*)d_in[1];
    const float* b1    = (const float*)d_in[2];
    const float* alpha = (const float*)d_in[3];
    const float* gam   = (const float*)d_in[4];
    const float* beta  = (const float*)d_in[5];
    const float* W_rep = (const float*)d_in[6];
    const float* vals  = (const float*)d_in[7];
    const float* sbias = (const float*)d_in[8];
    const int*   rows  = (const int*)d_in[9];
    const int*   cols  = (const int*)d_in[10];
    float* out = (float*)d_out;

    // workspace layout: [row_ptr (2501 ints, padded to 16KB)] [hA 640000 f32] [hB 640000 f32]
    int*   row_ptr = (int*)d_ws;
    float* hA = (float*)((char*)d_ws + 16384);
    float* hB = hA + (size_t)BATCH * VSZ;

    const int elems = BATCH * VSZ;
    const int eblk  = (elems + 255) / 256;

    rowptr_kernel<<<(NNODES + 1 + 255) / 256, 256, 0, stream>>>(rows, row_ptr);
    init_dense_kernel<<<eblk, 256, 0, stream>>>(x, W1, b1, hA);

    for (int i = 0; i < NLAYERS; ++i) {
        const int has_res = (i % 5 == 4) ? 1 : 0;
        const float* Wr = W_rep + (size_t)(i / 5) * 4 * VSZ;
        act_kernel<<<eblk, 256, 0, stream>>>(hA, hB,
                                             alpha + (size_t)i * VSZ,
                                             gam   + (size_t)i * VSZ,
                                             beta  + (size_t)i * VSZ,
                                             x, Wr, has_res);
        spmm_wmma_kernel<true><<<NNODES, 128, 0, stream>>>(
            vals + (size_t)i * NEDGES * (KB * KB),
            sbias + (size_t)i * VSZ,
            hB, row_ptr, cols, hA);
    }

    // final output sparse layer -> d_out in reference [BATCH][VSZ] layout
    spmm_wmma_kernel<false><<<NNODES, 128, 0, stream>>>(
        vals + (size_t)NLAYERS * NEDGES * (KB * KB),
        sbias + (size_t)NLAYERS * VSZ,
        hA, row_ptr, cols, out);
}